// PairList_4904852652341
// MI455X (gfx1250) — compile-verified
//
#include <hip/hip_runtime.h>
#include <cstdint>
#include <cstddef>

// Problem constants (match the reference generator)
#define NATOMS_PER_MOL 1024   // N
#define NDIM           3      // D
#define BLOCK          256    // threads per block (8 waves on wave32)
#define CHUNK          4096   // pairs handled per block (16 per thread)

// One molecule's coordinate slab: 1024 atoms * 3 floats * 4 B = 12288 B
#define SLAB_XFERS   ((NATOMS_PER_MOL * NDIM * 4) / 16)   // 768 x 16-byte async copies
#define XFER_ITERS   (SLAB_XFERS / BLOCK)                 // exactly 3, no remainder

typedef __attribute__((address_space(3))) float lds_f;

__device__ __forceinline__ void process_pair(
    int p, int npairs, int b0,
    const lds_f* __restrict__ sp,
    const float* __restrict__ coords,
    const int*   __restrict__ idx,
    float* __restrict__ out_i, float* __restrict__ out_j,
    float* __restrict__ out_d, float* __restrict__ out_r)
{
    // Streaming index reads: non-temporal (read-once).
    const int i = __builtin_nontemporal_load(idx + p);
    const int j = __builtin_nontemporal_load(idx + npairs + p);

    // Unconditional LDS gather (masked index is always in range); rare
    // molecule-straddling pairs are patched from global (hits L2: positions
    // are only 1.5 MB total).
    float xi, yi, zi, xj, yj, zj;
    {
        const int a = (i & (NATOMS_PER_MOL - 1)) * NDIM;
        xi = sp[a]; yi = sp[a + 1]; zi = sp[a + 2];
        if (__builtin_expect((i >> 10) != b0, 0)) {
            const float* g = coords + (size_t)i * NDIM;
            xi = g[0]; yi = g[1]; zi = g[2];
        }
    }
    {
        const int a = (j & (NATOMS_PER_MOL - 1)) * NDIM;
        xj = sp[a]; yj = sp[a + 1]; zj = sp[a + 2];
        if (__builtin_expect((j >> 10) != b0, 0)) {
            const float* g = coords + (size_t)j * NDIM;
            xj = g[0]; yj = g[1]; zj = g[2];
        }
    }

    const float rx = xi - xj;
    const float ry = yi - yj;
    const float rz = zi - zj;
    const float d  = sqrtf(rx * rx + ry * ry + rz * rz);

    // Write-once output streams: non-temporal stores (32-bit offsets).
    __builtin_nontemporal_store((float)i, out_i + p);
    __builtin_nontemporal_store((float)j, out_j + p);
    __builtin_nontemporal_store(d,        out_d + p);
    const int rb = 3 * p;                 // < 24M, fits int
    __builtin_nontemporal_store(rx, out_r + rb);
    __builtin_nontemporal_store(ry, out_r + rb + 1);
    __builtin_nontemporal_store(rz, out_r + rb + 2);
}

__global__ __launch_bounds__(BLOCK)
void pairlist_gfx1250_kernel(const float* __restrict__ coords,  // [B*N, 3]
                             const int*   __restrict__ idx,     // [2, npairs]
                             float*       __restrict__ out,     // [6*npairs]
                             int npairs)
{
    __shared__ __attribute__((aligned(16))) float spos[NATOMS_PER_MOL * NDIM];

    const int tid        = threadIdx.x;
    const int chunkStart = (int)blockIdx.x * CHUNK;   // <= 8M, fits int
    if (chunkStart >= npairs) return;

    // Pairs are emitted molecule-by-molecule: the first pair's molecule id
    // covers (nearly) the whole chunk.
    const int b0        = idx[chunkStart] >> 10;      // i / 1024 (uniform -> s_load)
    const int base_atom = b0 << 10;

    // Stage the molecule's 12 KB coordinate slab into LDS with CDNA5 async
    // global->LDS DMA (ASYNCcnt path, no VGPR round-trip). 768 transfers =
    // exactly 3 per thread -> branch-free.
    lds_f* sp = (lds_f*)spos;
    {
        const float*   gsrc     = coords + (size_t)base_atom * NDIM;
        const unsigned lds_base = (unsigned)(uintptr_t)sp;  // AS(3) ptr == byte offset
#pragma unroll
        for (int t = 0; t < XFER_ITERS; ++t) {
            const int      e    = tid + t * BLOCK;          // 0..767
            const unsigned ldsa = lds_base + (unsigned)e * 16u;
            const float*   g    = gsrc + (size_t)e * 4;
            asm volatile("global_load_async_to_lds_b128 %0, %1, off"
                         :: "v"(ldsa), "v"(g)
                         : "memory");
        }
        asm volatile("s_wait_asynccnt 0" ::: "memory");
    }
    __syncthreads();

    float* __restrict__ out_i = out;                   // idx row 0 (as f32)
    float* __restrict__ out_j = out +     npairs;      // idx row 1 (as f32)
    float* __restrict__ out_d = out + 2 * npairs;      // d_ij
    float* __restrict__ out_r = out + 3 * npairs;      // r_ij [npairs,3]

    if (chunkStart + CHUNK <= npairs) {
        // Fast path: full chunk, no bounds checks.
#pragma unroll
        for (int k = 0; k < CHUNK / BLOCK; ++k) {
            process_pair(chunkStart + tid + k * BLOCK, npairs, b0,
                         sp, coords, idx, out_i, out_j, out_d, out_r);
        }
    } else {
        // Tail path: only the final block takes this.
        for (int k = 0; k < CHUNK / BLOCK; ++k) {
            const int p = chunkStart + tid + k * BLOCK;
            if (p < npairs) {
                process_pair(p, npairs, b0,
                             sp, coords, idx, out_i, out_j, out_d, out_r);
            }
        }
    }
}

extern "C" void kernel_launch(void* const* d_in, const int* in_sizes, int n_in,
                              void* d_out, int out_size, void* d_ws, size_t ws_size,
                              hipStream_t stream) {
    const float* coords = (const float*)d_in[0];   // positions [B, N, 3] f32
    const int*   idx    = (const int*)d_in[1];     // atom_index12 [2, npairs] i32
    float*       out    = (float*)d_out;           // 6 * npairs floats

    const int npairs  = in_sizes[1] / 2;
    const int nblocks = (npairs + CHUNK - 1) / CHUNK;

    hipLaunchKernelGGL(pairlist_gfx1250_kernel, dim3(nblocks), dim3(BLOCK), 0,
                       stream, coords, idx, out, npairs);
}